// SkipGram_62989990363607
// MI455X (gfx1250) — compile-verified
//
#include <hip/hip_runtime.h>
#include <math.h>

#define N_NODES 200000
#define FEAT 128
#define K1_BLOCKS 1024
#define N_TILES (N_NODES / 16)   // 12500 exact

typedef float v2f __attribute__((ext_vector_type(2)));
typedef float v8f __attribute__((ext_vector_type(8)));

// Numerically safe (max, sumexp) merge: (m,s) <- combine((m,s),(m2,s2))
__device__ __forceinline__ void sm_combine(float& m, float& s, float m2, float s2) {
    float mn = fmaxf(m, m2);
    if (mn == -INFINITY) { m = mn; s = 0.0f; return; }     // both empty
    float a = (m  == -INFINITY) ? 0.0f : s  * expf(m  - mn);
    float b = (m2 == -INFINITY) ? 0.0f : s2 * expf(m2 - mn);
    m = mn; s = a + b;
}

// K1: z2 = z @ ctx[:, node] via V_WMMA_F32_16X16X4_F32, fused online-softmax partials.
__global__ __launch_bounds__(256) void gemv_wmma_kernel(
    const int* __restrict__ node_p,
    const float* __restrict__ z,          // [200000 x 128] row-major
    const float* __restrict__ ctx,        // [128 x 200000] row-major
    float* __restrict__ z2out,            // [200000] (d_out, holds z2 until K3)
    float* __restrict__ pm,               // [K1_BLOCKS] partial max
    float* __restrict__ ps)               // [K1_BLOCKS] partial sumexp
{
    __shared__ float z1s[FEAT];
    __shared__ float red_m[256];
    __shared__ float red_s[256];

    const int tid  = threadIdx.x;
    const int node = node_p[0];

    // Gather column `node` of ctx (stride N_NODES) into LDS once per block.
    for (int k = tid; k < FEAT; k += 256)
        z1s[k] = ctx[(size_t)k * N_NODES + node];
    __syncthreads();

    const int lane = tid & 31;
    const int wave = tid >> 5;
    const int hi   = lane >> 4;      // ISA A/B 16x4 layout: lanes 0-15 -> K{0,1}, 16-31 -> K{2,3}
    const int mrow = lane & 15;      // A-fragment row within 16-row tile
    const int koff = 2 * hi;

    float m = -INFINITY, s = 0.0f;

    const int gwave  = blockIdx.x * 8 + wave;
    const int nwaves = gridDim.x * 8;

    for (int t = gwave; t < N_TILES; t += nwaves) {
        const float* arow = z + (size_t)(t * 16 + mrow) * FEAT + koff;
        v8f c0 = {};
        v8f c1 = {};
        #pragma unroll
        for (int k = 0; k < FEAT; k += 8) {
            v2f a0 = *(const v2f*)(arow + k);
            v2f a1 = *(const v2f*)(arow + k + 4);
            v2f b0 = *(const v2f*)(&z1s[k + koff]);
            v2f b1 = *(const v2f*)(&z1s[k + 4 + koff]);
            // D = A(16x4,f32) x B(4x16,f32) + C ; B column-broadcast of z1 chunk
            c0 = __builtin_amdgcn_wmma_f32_16x16x4_f32(
                     false, a0, false, b0, (short)0, c0, false, false);
            c1 = __builtin_amdgcn_wmma_f32_16x16x4_f32(
                     false, a1, false, b1, (short)0, c1, false, false);
        }
        v8f c = c0 + c1;

        // C/D layout: VGPR j holds row (j + 8*hi); all 16 columns identical.
        // Lanes 0-7 extract rows 0-7, lanes 16-23 extract rows 8-15.
        if (mrow < 8) {
            int r = mrow;
            float d0 = (r & 1) ? c[1] : c[0];
            float d1 = (r & 1) ? c[3] : c[2];
            float d2 = (r & 1) ? c[5] : c[4];
            float d3 = (r & 1) ? c[7] : c[6];
            float e0 = (r & 2) ? d1 : d0;
            float e1 = (r & 2) ? d3 : d2;
            float d  = (r & 4) ? e1 : e0;

            int row = t * 16 + 8 * hi + r;
            z2out[row] = d;

            // online logsumexp update (m=-inf first pass: expf(-inf)=0)
            float mn = fmaxf(m, d);
            s = s * expf(m - mn) + expf(d - mn);
            m = mn;
        }
    }

    // Block tree-reduce of (m, s) pairs.
    red_m[tid] = m; red_s[tid] = s;
    __syncthreads();
    for (int off = 128; off > 0; off >>= 1) {
        if (tid < off) {
            sm_combine(m, s, red_m[tid + off], red_s[tid + off]);
            red_m[tid] = m; red_s[tid] = s;
        }
        __syncthreads();
    }
    if (tid == 0) { pm[blockIdx.x] = m; ps[blockIdx.x] = s; }
}

// K2: combine all block partials -> c = global_max + log(sum exp(z2 - global_max))
__global__ __launch_bounds__(256) void logsumexp_reduce_kernel(
    const float* __restrict__ pm, const float* __restrict__ ps,
    float* __restrict__ cout)
{
    __shared__ float red_m[256];
    __shared__ float red_s[256];
    const int tid = threadIdx.x;

    float m = -INFINITY, s = 0.0f;
    for (int i = tid; i < K1_BLOCKS; i += 256)
        sm_combine(m, s, pm[i], ps[i]);

    red_m[tid] = m; red_s[tid] = s;
    __syncthreads();
    for (int off = 128; off > 0; off >>= 1) {
        if (tid < off) {
            sm_combine(m, s, red_m[tid + off], red_s[tid + off]);
            red_m[tid] = m; red_s[tid] = s;
        }
        __syncthreads();
    }
    if (tid == 0) cout[0] = m + logf(s);
}

// K3: out[i] = z2[i] - c
__global__ void finalize_kernel(float* __restrict__ out,
                                const float* __restrict__ cptr, int n)
{
    int i = blockIdx.x * blockDim.x + threadIdx.x;
    if (i < n) out[i] = out[i] - cptr[0];
}

extern "C" void kernel_launch(void* const* d_in, const int* in_sizes, int n_in,
                              void* d_out, int out_size, void* d_ws, size_t ws_size,
                              hipStream_t stream) {
    const int*   node = (const int*)d_in[0];
    const float* z    = (const float*)d_in[1];
    const float* ctx  = (const float*)d_in[2];
    float* out = (float*)d_out;

    float* pm   = (float*)d_ws;
    float* ps   = pm + K1_BLOCKS;
    float* cbuf = ps + K1_BLOCKS;

    gemv_wmma_kernel<<<K1_BLOCKS, 256, 0, stream>>>(node, z, ctx, out, pm, ps);
    logsumexp_reduce_kernel<<<1, 256, 0, stream>>>(pm, ps, cbuf);
    finalize_kernel<<<(N_NODES + 255) / 256, 256, 0, stream>>>(out, cbuf, N_NODES);
}